// myGATConv_85375359910105
// MI455X (gfx1250) — compile-verified
//
#include <hip/hip_runtime.h>
#include <hip/hip_bf16.h>

// ---------------- problem constants (match reference) ----------------
#define NN 50000
#define EE 800000
#define INF_ 128      // in_feats
#define HH 4          // heads
#define DD 32         // out feats / head
#define EF_ 32        // edge feats
#define NEG_SLOPE 0.2f

typedef __attribute__((ext_vector_type(2))) float v2f;
typedef __attribute__((ext_vector_type(8))) float v8f;

// ---------------- helpers ----------------
__device__ __forceinline__ unsigned ford(float f) {
    unsigned b = __float_as_uint(f);
    return (b & 0x80000000u) ? ~b : (b | 0x80000000u);
}
__device__ __forceinline__ float fdec(unsigned o) {
    return __uint_as_float((o & 0x80000000u) ? (o & 0x7FFFFFFFu) : ~o);
}
__device__ __forceinline__ float lrelu(float x) {
    return x > 0.f ? x : NEG_SLOPE * x;
}

// ---------------- kernel 1: 8x4 edge-type attention table ----------------
// ee[t,h] = sum_f ( sum_k W_e[h*32+f,k] * t*edge_emb[t,k] ) * attn_e[t,h,f]
__global__ void gat_ee_table(const float* __restrict__ W_e,
                             const float* __restrict__ edge_emb,
                             const float* __restrict__ attn_e,
                             float* __restrict__ ee_table) {
    int t = threadIdx.x;              // 0..31
    if (t >= 32) return;
    int et = t >> 2;                  // 0..7
    int h  = t & 3;                   // 0..3
    const float* em = edge_emb + et * EF_;
    float acc = 0.f;
    for (int f = 0; f < EF_; ++f) {
        const float* wrow = W_e + (h * EF_ + f) * EF_;
        float p = 0.f;
        for (int k = 0; k < EF_; ++k) p += wrow[k] * em[k];
        acc += ((float)et * p) * attn_e[(et * HH + h) * EF_ + f];
    }
    ee_table[et * HH + h] = acc;
}

// ---------------- kernel 2: feat_src = feat @ W_fc.T via f32 WMMA ----------------
// Grid: 3125 blocks (M tiles of 16), block = 256 threads = 8 wave32s.
// Wave w computes the 16x16 output tile at columns [16w, 16w+16).
#define LDA 132   // padded LDS row stride (floats); 132*4B = 528B, 16B aligned, bank-spread
__global__ __launch_bounds__(256)
void gat_proj_wmma(const float* __restrict__ feat,
                   const float* __restrict__ W_fc,
                   float* __restrict__ feat_src) {
    __shared__ float As[16 * LDA];
    const int row0 = blockIdx.x * 16;
    const int tid  = threadIdx.x;

    // cooperative load of A tile (16 x 128 f32) into LDS, stride-padded
    {
        int r   = tid >> 4;   // 0..15
        int seg = tid & 15;   // 0..15 -> 8 floats each
        const float4* s4 = (const float4*)(feat + (size_t)(row0 + r) * INF_ + seg * 8);
        float4 p0 = s4[0];
        float4 p1 = s4[1];
        float4* d4 = (float4*)(As + r * LDA + seg * 8);
        d4[0] = p0;
        d4[1] = p1;
    }
    __syncthreads();

    const int wave  = tid >> 5;        // 0..7 -> N tile
    const int lane  = tid & 31;
    const int n0    = wave * 16;
    const int lhalf = lane >> 4;       // 0: K pair {0,1}, 1: K pair {2,3}
    const int lmod  = lane & 15;       // M (for A) / N (for B) within tile

    v8f c = {};
    #pragma unroll 4
    for (int kk = 0; kk < INF_ / 4; ++kk) {
        int kb = 4 * kk + 2 * lhalf;
        // A fragment: row = lmod, K = kb..kb+1 (ISA 7.12.2 32-bit 16x4 layout)
        v2f a = *(const v2f*)(As + lmod * LDA + kb);
        // B fragment: B[k][n] = W_fc[n][k]; N = n0+lmod, K = kb..kb+1
        v2f b = *(const v2f*)(W_fc + (size_t)(n0 + lmod) * INF_ + kb);
        c = __builtin_amdgcn_wmma_f32_16x16x4_f32(
                /*neg_a=*/false, a, /*neg_b=*/false, b,
                /*c_mod=*/(short)0, c, /*reuse_a=*/false, /*reuse_b=*/false);
    }

    // C layout: VGPR g -> M = g + 8*lhalf, N = lmod
    float* out = feat_src + (size_t)row0 * (HH * DD) + n0;
    #pragma unroll
    for (int g = 0; g < 8; ++g) {
        int m = g + 8 * lhalf;
        out[(size_t)m * (HH * DD) + lmod] = c[g];
    }
}

// ---------------- kernel 3: el/er per (node, head) ----------------
// 128 threads per node; each wave32 is one head; shuffle reduction.
__global__ __launch_bounds__(128)
void gat_node_attn(const float* __restrict__ feat_src,
                   const float* __restrict__ attn_l,
                   const float* __restrict__ attn_r,
                   const int* __restrict__ ntype,
                   float* __restrict__ el, float* __restrict__ er) {
    int n = blockIdx.x;
    int t = threadIdx.x;              // h = t>>5, d = t&31
    int nt = ntype[n];
    float fs = feat_src[(size_t)n * (HH * DD) + t];
    float vl = fs * attn_l[nt * (HH * DD) + t];
    float vr = fs * attn_r[nt * (HH * DD) + t];
    #pragma unroll
    for (int off = 16; off > 0; off >>= 1) {
        vl += __shfl_xor(vl, off, 32);
        vr += __shfl_xor(vr, off, 32);
    }
    if ((t & 31) == 0) {
        int h = t >> 5;
        el[n * HH + h] = vl;
        er[n * HH + h] = vr;
    }
}

// ---------------- kernel 4: edge scores + segment max (ordered-uint atomic) ----------------
__global__ __launch_bounds__(256)
void gat_edge_score(const int* __restrict__ src, const int* __restrict__ dst,
                    const int* __restrict__ etype,
                    const float* __restrict__ el, const float* __restrict__ er,
                    const float* __restrict__ ee_table,
                    float* __restrict__ e_buf, unsigned* __restrict__ m_ord) {
    int i = blockIdx.x * blockDim.x + threadIdx.x;
    if (i >= EE) return;
    int s = src[i], d = dst[i], t = etype[i];
    float4 lv = *(const float4*)(el + (size_t)s * HH);
    float4 rv = *(const float4*)(er + (size_t)d * HH);
    float4 tv = *(const float4*)(ee_table + t * HH);
    float4 ev;
    ev.x = lrelu(lv.x + rv.x + tv.x);
    ev.y = lrelu(lv.y + rv.y + tv.y);
    ev.z = lrelu(lv.z + rv.z + tv.z);
    ev.w = lrelu(lv.w + rv.w + tv.w);
    *(float4*)(e_buf + (size_t)i * HH) = ev;
    unsigned* mp = m_ord + (size_t)d * HH;
    atomicMax(mp + 0, ford(ev.x));
    atomicMax(mp + 1, ford(ev.y));
    atomicMax(mp + 2, ford(ev.z));
    atomicMax(mp + 3, ford(ev.w));
}

// ---------------- kernel 5: exp(e - m[dst]) + segment sum ----------------
__global__ __launch_bounds__(256)
void gat_edge_expsum(const int* __restrict__ dst,
                     float* __restrict__ e_buf,
                     const unsigned* __restrict__ m_ord,
                     float* __restrict__ den) {
    int i = blockIdx.x * blockDim.x + threadIdx.x;
    if (i >= EE) return;
    int d = dst[i];
    const unsigned* mp = m_ord + (size_t)d * HH;
    float4 ev = *(const float4*)(e_buf + (size_t)i * HH);
    float4 xv;
    xv.x = expf(ev.x - fdec(mp[0]));
    xv.y = expf(ev.y - fdec(mp[1]));
    xv.z = expf(ev.z - fdec(mp[2]));
    xv.w = expf(ev.w - fdec(mp[3]));
    *(float4*)(e_buf + (size_t)i * HH) = xv;
    float* dp = den + (size_t)d * HH;
    atomicAdd(dp + 0, xv.x);
    atomicAdd(dp + 1, xv.y);
    atomicAdd(dp + 2, xv.z);
    atomicAdd(dp + 3, xv.w);
}

// ---------------- kernel 6: normalize + aggregate (one wave32 per edge) ----------------
__global__ __launch_bounds__(256)
void gat_norm_agg(const int* __restrict__ src, const int* __restrict__ dst,
                  const float* __restrict__ ex_buf, const float* __restrict__ den,
                  const float* __restrict__ feat_src,
                  float* __restrict__ rst, float* __restrict__ out_a) {
    int gid  = blockIdx.x * blockDim.x + threadIdx.x;
    int edge = gid >> 5;
    int lane = gid & 31;
    if (edge >= EE) return;          // whole wave exits together (wave-aligned)
    int s = src[edge], d = dst[edge];

    float aval = 0.f;
    if (lane < HH) {
        aval = ex_buf[(size_t)edge * HH + lane] / den[(size_t)d * HH + lane];
        out_a[(size_t)edge * HH + lane] = aval;
    }
    float a0 = __shfl(aval, 0, 32);
    float a1 = __shfl(aval, 1, 32);
    float a2 = __shfl(aval, 2, 32);
    float a3 = __shfl(aval, 3, 32);

    const float* fs = feat_src + (size_t)s * (HH * DD);
    float* rd = rst + (size_t)d * (HH * DD);
    atomicAdd(&rd[0 * DD + lane], a0 * fs[0 * DD + lane]);
    atomicAdd(&rd[1 * DD + lane], a1 * fs[1 * DD + lane]);
    atomicAdd(&rd[2 * DD + lane], a2 * fs[2 * DD + lane]);
    atomicAdd(&rd[3 * DD + lane], a3 * fs[3 * DD + lane]);
}

// ---------------- host launch ----------------
extern "C" void kernel_launch(void* const* d_in, const int* in_sizes, int n_in,
                              void* d_out, int out_size, void* d_ws, size_t ws_size,
                              hipStream_t stream) {
    const float* feat     = (const float*)d_in[0];
    const float* W_fc     = (const float*)d_in[1];
    const float* W_e      = (const float*)d_in[2];
    const float* edge_emb = (const float*)d_in[3];
    const float* attn_l   = (const float*)d_in[4];
    const float* attn_r   = (const float*)d_in[5];
    const float* attn_e   = (const float*)d_in[6];
    const int*   src      = (const int*)d_in[7];
    const int*   dst      = (const int*)d_in[8];
    const int*   etype    = (const int*)d_in[9];
    const int*   ntype    = (const int*)d_in[10];

    float* out   = (float*)d_out;
    float* rst   = out;                                  // (N, H, D)
    float* out_a = out + (size_t)NN * HH * DD;           // (E, H, 1)

    // workspace layout (floats); total ~10.4M floats (~41.6 MB)
    float* ws = (float*)d_ws;
    size_t off = 0;
    float*    feat_src = ws + off; off += (size_t)NN * HH * DD;  // 6.4M
    float*    el       = ws + off; off += (size_t)NN * HH;       // 200K
    float*    er       = ws + off; off += (size_t)NN * HH;       // 200K
    float*    ee_tab   = ws + off; off += 64;
    unsigned* m_ord    = (unsigned*)(ws + off); off += (size_t)NN * HH;
    float*    den      = ws + off; off += (size_t)NN * HH;
    float*    e_buf    = ws + off; off += (size_t)EE * HH;       // 3.2M

    // zero the accumulators every call (harness does not re-poison between replays)
    hipMemsetAsync(m_ord, 0, (size_t)NN * HH * sizeof(unsigned), stream);
    hipMemsetAsync(den,   0, (size_t)NN * HH * sizeof(float),    stream);
    hipMemsetAsync(rst,   0, (size_t)NN * HH * DD * sizeof(float), stream);

    gat_ee_table<<<1, 32, 0, stream>>>(W_e, edge_emb, attn_e, ee_tab);
    gat_proj_wmma<<<NN / 16, 256, 0, stream>>>(feat, W_fc, feat_src);
    gat_node_attn<<<NN, 128, 0, stream>>>(feat_src, attn_l, attn_r, ntype, el, er);
    gat_edge_score<<<(EE + 255) / 256, 256, 0, stream>>>(src, dst, etype, el, er,
                                                         ee_tab, e_buf, m_ord);
    gat_edge_expsum<<<(EE + 255) / 256, 256, 0, stream>>>(dst, e_buf, m_ord, den);
    gat_norm_agg<<<((size_t)EE * 32 + 255) / 256, 256, 0, stream>>>(src, dst, e_buf, den,
                                                                    feat_src, rst, out_a);
    (void)in_sizes; (void)n_in; (void)out_size; (void)ws_size;
}